// MemoryCompressedAttention_64682207477865
// MI455X (gfx1250) — compile-verified
//
#include <hip/hip_runtime.h>

typedef __attribute__((ext_vector_type(2))) float v2f;
typedef __attribute__((ext_vector_type(8))) float v8f;

#define B_    2
#define L_    4096
#define D_    512
#define H_    8
#define R_    3
#define PAD_  2          // (-L) % R
#define COMP_ 1366       // (L + PAD) / R
#define DK_   64
#define CF_   3          // ceil(L / COMP)

// fp32 WMMA: D(16x16,f32) = A(16x4,f32) * B(4x16,f32) + C
// fragment layout (wave32): half = lane>>4, l16 = lane&15
//   A: row m = l16,  k = k0 + 2*half + vg   (vg = vector component)
//   B: col n = l16,  k = k0 + 2*half + vg
//   C/D: row m = v + 8*half, col n = l16
static __device__ __forceinline__ v8f wmma4(v2f a, v2f b, v8f c) {
  return __builtin_amdgcn_wmma_f32_16x16x4_f32(false, a, false, b, (short)0, c,
                                               false, false);
}

// ---------------------------------------------------------------------------
// Kernel 0: repack conv weights Wc(o, d, k) -> Wc'(o, k*512 + d) so the
// compress GEMM's B-fragments are contiguous b64 loads along its K axis.
// ---------------------------------------------------------------------------
__global__ __launch_bounds__(256) void nsa_repack_wc(
    const float* __restrict__ Wc, float* __restrict__ Wcp) {
  int idx = blockIdx.x * 256 + threadIdx.x;   // < 512*1536
  int n  = idx / (R_ * D_);
  int kk = idx - n * (R_ * D_);
  int tap = kk >> 9;
  int d   = kk & 511;
  Wcp[idx] = Wc[(n * D_ + d) * R_ + tap];
}

// ---------------------------------------------------------------------------
// Kernel 1: compressed K/V = strided conv1d (kernel=stride=R) as implicit
// im2col GEMM: out[(b,t),o] = sum_{tap,d} pad(src)[b,3t+tap,d]*Wc'[o,tap*512+d] + bc
// A rows gathered on the fly; front-pad handled by clamped b64 load * {0,1}.
// Inner loop: 10-load clause, then 8 WMMAs (K unrolled by 8).
// ---------------------------------------------------------------------------
__global__ __launch_bounds__(256) void nsa_compress(
    const float* __restrict__ key, const float* __restrict__ value,
    const float* __restrict__ Wcp, const float* __restrict__ bc,
    float* __restrict__ kc, float* __restrict__ vc) {
  const int lane = threadIdx.x & 31;
  const int wave = threadIdx.x >> 5;
  const int half = lane >> 4;
  const int l16  = lane & 15;

  const float* __restrict__ src = blockIdx.z ? value : key;
  float* __restrict__ dst       = blockIdx.z ? vc : kc;

  const int M        = B_ * COMP_;        // 2732
  const int rowTiles = (M + 15) >> 4;     // 171
  const int colTiles = D_ / 64;           // 8
  const int tile = blockIdx.x * 8 + wave;
  if (tile >= rowTiles * colTiles) return;
  const int rowT = tile / colTiles;
  const int colT = tile % colTiles;
  const int m0 = rowT * 16;
  const int n0 = colT * 64;

  int m = m0 + l16;
  if (m >= M) m = M - 1;               // ragged last tile: stores are guarded
  const int bi = m / COMP_;
  const int t  = m - bi * COMP_;

  const float* __restrict__ aptr[R_];
  float aval[R_];
#pragma unroll
  for (int tap = 0; tap < R_; ++tap) {
    int srow = 3 * t + tap - PAD_;
    aval[tap] = (srow >= 0 && srow < L_) ? 1.0f : 0.0f;
    int srowc = srow < 0 ? 0 : (srow >= L_ ? L_ - 1 : srow);
    aptr[tap] = src + ((long)bi * L_ + srowc) * D_;
  }

  const float* __restrict__ wrow[4];
#pragma unroll
  for (int tt = 0; tt < 4; ++tt)
    wrow[tt] = Wcp + (long)(n0 + tt * 16 + l16) * (R_ * D_);

  v8f acc[4];
#pragma unroll
  for (int tt = 0; tt < 4; ++tt) {
    float bb = bc[n0 + tt * 16 + l16];
#pragma unroll
    for (int v = 0; v < 8; ++v) acc[tt][v] = bb;
  }

#pragma unroll
  for (int tap = 0; tap < R_; ++tap) {
    const float* __restrict__ ap = aptr[tap];
    const float  av = aval[tap];
    const int kb = tap * D_;
    for (int kk0 = 0; kk0 < D_; kk0 += 8) {
      // ---- load phase: 2 A frags + 8 B frags --------------------------
      v2f a0 = *(const v2f*)(ap + kk0 + 2 * half);
      v2f a1 = *(const v2f*)(ap + kk0 + 4 + 2 * half);
      v2f b0[4], b1[4];
#pragma unroll
      for (int tt = 0; tt < 4; ++tt)
        b0[tt] = *(const v2f*)(wrow[tt] + kb + kk0 + 2 * half);
#pragma unroll
      for (int tt = 0; tt < 4; ++tt)
        b1[tt] = *(const v2f*)(wrow[tt] + kb + kk0 + 4 + 2 * half);
      a0.x *= av; a0.y *= av;
      a1.x *= av; a1.y *= av;
      // ---- compute phase: 8 WMMAs -------------------------------------
#pragma unroll
      for (int tt = 0; tt < 4; ++tt) acc[tt] = wmma4(a0, b0[tt], acc[tt]);
#pragma unroll
      for (int tt = 0; tt < 4; ++tt) acc[tt] = wmma4(a1, b1[tt], acc[tt]);
    }
  }

#pragma unroll
  for (int tt = 0; tt < 4; ++tt)
#pragma unroll
    for (int v = 0; v < 8; ++v) {
      int mr = m0 + v + 8 * half;
      if (mr < M) dst[(long)mr * D_ + n0 + tt * 16 + l16] = acc[tt][v];
    }
}

// ---------------------------------------------------------------------------
// Kernel 2: generic projection  C(M,512) = A(M,512) @ W(512,512)^T + bias
// Each wave: 16 rows x 64 cols (4 accumulators), A fragment reused 4x.
// Inner loop: 10-load clause, then 8 WMMAs (K unrolled by 8).
// ---------------------------------------------------------------------------
__global__ __launch_bounds__(256) void nsa_proj(
    const float* __restrict__ A, const float* __restrict__ W,
    const float* __restrict__ bias, float* __restrict__ C, int M) {
  const int lane = threadIdx.x & 31;
  const int wave = threadIdx.x >> 5;
  const int half = lane >> 4;
  const int l16  = lane & 15;

  const int rowTiles = (M + 15) >> 4;
  const int colTiles = D_ / 64;
  const int tile = blockIdx.x * 8 + wave;
  if (tile >= rowTiles * colTiles) return;
  const int rowT = tile / colTiles;
  const int colT = tile % colTiles;
  const int m0 = rowT * 16;
  const int n0 = colT * 64;

  int m = m0 + l16;
  if (m >= M) m = M - 1;
  const float* __restrict__ arow = A + (long)m * D_;

  const float* __restrict__ wrow[4];
#pragma unroll
  for (int tt = 0; tt < 4; ++tt)
    wrow[tt] = W + (long)(n0 + tt * 16 + l16) * D_;

  v8f acc[4];
#pragma unroll
  for (int tt = 0; tt < 4; ++tt) {
    float bb = bias[n0 + tt * 16 + l16];
#pragma unroll
    for (int v = 0; v < 8; ++v) acc[tt][v] = bb;
  }

  for (int kk0 = 0; kk0 < D_; kk0 += 8) {
    v2f a0 = *(const v2f*)(arow + kk0 + 2 * half);
    v2f a1 = *(const v2f*)(arow + kk0 + 4 + 2 * half);
    v2f b0[4], b1[4];
#pragma unroll
    for (int tt = 0; tt < 4; ++tt)
      b0[tt] = *(const v2f*)(wrow[tt] + kk0 + 2 * half);
#pragma unroll
    for (int tt = 0; tt < 4; ++tt)
      b1[tt] = *(const v2f*)(wrow[tt] + kk0 + 4 + 2 * half);
#pragma unroll
    for (int tt = 0; tt < 4; ++tt) acc[tt] = wmma4(a0, b0[tt], acc[tt]);
#pragma unroll
    for (int tt = 0; tt < 4; ++tt) acc[tt] = wmma4(a1, b1[tt], acc[tt]);
  }

#pragma unroll
  for (int tt = 0; tt < 4; ++tt)
#pragma unroll
    for (int v = 0; v < 8; ++v) {
      int mr = m0 + v + 8 * half;
      if (mr < M) C[(long)mr * D_ + n0 + tt * 16 + l16] = acc[tt][v];
    }
}

// ---------------------------------------------------------------------------
// Kernel 3: flash-style attention over compressed keys.
// Wave tile = 16 queries; S = (Q*scale) K^T, online softmax with shfl_xor row
// reductions (C rows live in 16-lane halves), P routed through a 17-padded
// private LDS tile to get A-layout, O += P V. All fragment loads for a key
// tile are batched before the WMMA bursts. Causal mask bounds the loop.
// ---------------------------------------------------------------------------
__global__ __launch_bounds__(256) void nsa_attn(
    const float* __restrict__ q, const float* __restrict__ kh,
    const float* __restrict__ vh, float* __restrict__ o) {
  __shared__ float pbuf[8][16 * 17];
  const int lane = threadIdx.x & 31;
  const int wave = threadIdx.x >> 5;
  const int half = lane >> 4;
  const int l16  = lane & 15;
  const int b = blockIdx.z;
  const int h = blockIdx.y;
  const int q0 = blockIdx.x * 128 + wave * 16;
  const float scale = 0.125f;  // 1/sqrt(dk)

  // preload Q fragments for the whole dk=64 strip, scale folded in
  v2f qa[16];
  const float* __restrict__ qrow =
      q + ((long)(b * L_ + q0 + l16) * D_) + h * DK_;
#pragma unroll
  for (int ks = 0; ks < 16; ++ks) {
    v2f t = *(const v2f*)(qrow + ks * 4 + 2 * half);
    t.x *= scale;
    t.y *= scale;
    qa[ks] = t;
  }

  v8f O[4];
  float mrun[8], lrun[8];
#pragma unroll
  for (int tt = 0; tt < 4; ++tt)
#pragma unroll
    for (int v = 0; v < 8; ++v) O[tt][v] = 0.0f;
#pragma unroll
  for (int v = 0; v < 8; ++v) {
    mrun[v] = -1e30f;
    lrun[v] = 0.0f;
  }

  const int jmax   = (q0 + 15) / CF_;   // largest allowed compressed key
  const int ntiles = (jmax >> 4) + 1;

  const float* __restrict__ khb = kh + (long)b * COMP_ * D_ + h * DK_;
  const float* __restrict__ vhb = vh + (long)b * COMP_ * D_ + h * DK_;
  float* pl = &pbuf[wave][0];

  for (int kt = 0; kt < ntiles; ++kt) {
    // ---- scores: batch 16 K-fragment loads, then 16 WMMAs -------------
    const int j  = kt * 16 + l16;                 // this lane's key column
    const int jc = j < COMP_ ? j : COMP_ - 1;
    const float* __restrict__ krow = khb + (long)jc * D_;
    v2f kfr[16];
#pragma unroll
    for (int ks = 0; ks < 16; ++ks)
      kfr[ks] = *(const v2f*)(krow + ks * 4 + 2 * half);
    v8f S;
#pragma unroll
    for (int v = 0; v < 8; ++v) S[v] = 0.0f;
#pragma unroll
    for (int ks = 0; ks < 16; ++ks) S = wmma4(qa[ks], kfr[ks], S);

    // ---- mask + online softmax ----------------------------------------
#pragma unroll
    for (int v = 0; v < 8; ++v) {
      const int i = q0 + v + 8 * half;            // query index of this row
      const bool ok = (j <= i / CF_) && (j < COMP_);
      float s = ok ? S[v] : -1e30f;
      float r = s;
      r = fmaxf(r, __shfl_xor(r, 1, 32));
      r = fmaxf(r, __shfl_xor(r, 2, 32));
      r = fmaxf(r, __shfl_xor(r, 4, 32));
      r = fmaxf(r, __shfl_xor(r, 8, 32));
      const float mnew = fmaxf(mrun[v], r);
      const float corr = __expf(mrun[v] - mnew);
      const float p    = __expf(s - mnew);
      float rs = p;
      rs += __shfl_xor(rs, 1, 32);
      rs += __shfl_xor(rs, 2, 32);
      rs += __shfl_xor(rs, 4, 32);
      rs += __shfl_xor(rs, 8, 32);
      lrun[v] = lrun[v] * corr + rs;
      mrun[v] = mnew;
#pragma unroll
      for (int tt = 0; tt < 4; ++tt) O[tt][v] *= corr;
      pl[(v + 8 * half) * 17 + l16] = p;          // C-layout -> LDS
    }

    // ---- O += P V: batch 8 LDS + 16 global loads, then 16 WMMAs -------
    v2f pa[4];
    v2f vfr[4][4];
#pragma unroll
    for (int ks = 0; ks < 4; ++ks) {
      pa[ks].x = pl[l16 * 17 + ks * 4 + 2 * half + 0];
      pa[ks].y = pl[l16 * 17 + ks * 4 + 2 * half + 1];
      const float* __restrict__ vrow[2];
#pragma unroll
      for (int vg = 0; vg < 2; ++vg) {
        int jj  = kt * 16 + ks * 4 + 2 * half + vg;
        int jjc = jj < COMP_ ? jj : COMP_ - 1;    // masked => p==0 anyway
        vrow[vg] = vhb + (long)jjc * D_;
      }
#pragma unroll
      for (int tt = 0; tt < 4; ++tt) {
        vfr[ks][tt].x = vrow[0][tt * 16 + l16];
        vfr[ks][tt].y = vrow[1][tt * 16 + l16];
      }
    }
#pragma unroll
    for (int ks = 0; ks < 4; ++ks)
#pragma unroll
      for (int tt = 0; tt < 4; ++tt)
        O[tt] = wmma4(pa[ks], vfr[ks][tt], O[tt]);
  }

  // ---- epilogue: normalize and write O --------------------------------
#pragma unroll
  for (int v = 0; v < 8; ++v) {
    const float inv = 1.0f / lrun[v];
    const int i = q0 + v + 8 * half;
    float* __restrict__ orow = o + ((long)(b * L_ + i) * D_) + h * DK_;
#pragma unroll
    for (int tt = 0; tt < 4; ++tt)
      orow[tt * 16 + l16] = O[tt][v] * inv;
  }
}

// ---------------------------------------------------------------------------
extern "C" void kernel_launch(void* const* d_in, const int* in_sizes, int n_in,
                              void* d_out, int out_size, void* d_ws,
                              size_t ws_size, hipStream_t stream) {
  (void)in_sizes; (void)n_in; (void)out_size; (void)ws_size;
  const float* query = (const float*)d_in[0];
  const float* key   = (const float*)d_in[1];
  const float* value = (const float*)d_in[2];
  const float* Wq = (const float*)d_in[3];
  const float* bq = (const float*)d_in[4];
  const float* Wk = (const float*)d_in[5];
  const float* bk = (const float*)d_in[6];
  const float* Wv = (const float*)d_in[7];
  const float* bv = (const float*)d_in[8];
  const float* Wo = (const float*)d_in[9];
  const float* bo = (const float*)d_in[10];
  const float* Wc = (const float*)d_in[11];
  const float* bc = (const float*)d_in[12];
  float* out = (float*)d_out;

  const long MQ = (long)B_ * L_;     // 8192
  const long MC = (long)B_ * COMP_;  // 2732

  float* ws = (float*)d_ws;
  float* qbuf  = ws; ws += MQ * D_;
  float* obuf  = ws; ws += MQ * D_;
  float* kcbuf = ws; ws += MC * D_;
  float* vcbuf = ws; ws += MC * D_;
  float* khbuf = ws; ws += MC * D_;
  float* vhbuf = ws; ws += MC * D_;
  float* wcbuf = ws; ws += (long)D_ * R_ * D_;   // repacked conv weights

  // 0) repack conv weights to GEMM-K-contiguous layout
  nsa_repack_wc<<<dim3((D_ * R_ * D_) / 256), 256, 0, stream>>>(Wc, wcbuf);

  // 1) conv-compress K and V (grid.z selects K vs V)
  {
    int tiles = (((int)MC + 15) / 16) * (D_ / 64);
    nsa_compress<<<dim3((tiles + 7) / 8, 1, 2), 256, 0, stream>>>(
        key, value, wcbuf, bc, kcbuf, vcbuf);
  }
  // 2) projections: Q on full sequence, K/V heads on compressed sequence
  {
    int tilesQ = ((int)MQ / 16) * (D_ / 64);
    nsa_proj<<<dim3((tilesQ + 7) / 8), 256, 0, stream>>>(query, Wq, bq, qbuf,
                                                         (int)MQ);
    int tilesC = (((int)MC + 15) / 16) * (D_ / 64);
    nsa_proj<<<dim3((tilesC + 7) / 8), 256, 0, stream>>>(kcbuf, Wk, bk, khbuf,
                                                         (int)MC);
    nsa_proj<<<dim3((tilesC + 7) / 8), 256, 0, stream>>>(vcbuf, Wv, bv, vhbuf,
                                                         (int)MC);
  }
  // 3) block-causal flash attention over compressed keys
  nsa_attn<<<dim3(L_ / 128, H_, B_), 256, 0, stream>>>(qbuf, khbuf, vhbuf,
                                                       obuf);
  // 4) output projection straight into d_out
  {
    int tilesQ = ((int)MQ / 16) * (D_ / 64);
    nsa_proj<<<dim3((tilesQ + 7) / 8), 256, 0, stream>>>(obuf, Wo, bo, out,
                                                         (int)MQ);
  }
}